// MultiHeadLatentAttention_74517682586449
// MI455X (gfx1250) — compile-verified
//
#include <hip/hip_runtime.h>
#include <hip/hip_bf16.h>
#include <math.h>

typedef float v2f __attribute__((ext_vector_type(2)));
typedef float v8f __attribute__((ext_vector_type(8)));

#define BM 128
#define BN 64
#define BK 16
#define LDSS 20   // padded LDS row stride (floats): 80B -> 16B-aligned stores, conflict-free b64 reads

namespace cfg {
constexpr int B = 2, S = 2048, HIDDEN = 2048, H = 16;
constexpr int QL = 768, KVL = 512, NOPE = 128, ROPE = 64, VD = 128, QKH = 192;
constexpr int KC = KVL + ROPE; // 576 = concat(kv_latent, k_pe)
}

// Fragment load for V_WMMA_F32_16X16X4_F32.
// A 16x4 f32 layout (ISA 7.12.2): lanes 0-15 hold M=lane, K={0,1}; lanes 16-31 hold M=lane-16, K={2,3}.
// B 4x16 mirrors with N<->M; staging A[m][k] and B^T[n][k] identically makes both loads the same.
__device__ inline v2f lds_frag(const float* s, int rowBase, int kk, int lane) {
  const int r = rowBase + (lane & 15);
  const int k = kk + ((lane >> 4) << 1);
  const float2 t = *(const float2*)(s + r * LDSS + k);
  v2f v; v.x = t.x; v.y = t.y;
  return v;
}

__device__ inline v8f wmma4(v2f a, v2f b, v8f c) {
  return __builtin_amdgcn_wmma_f32_16x16x4_f32(false, a, false, b, (short)0, c, false, false);
}

// Generic batched C[MxN] = A[MxK] * B[NxK]^T, f32 WMMA 16x16x4.
// 256 threads = 8 waves (4 along M x 2 along N); each wave owns a 32x32 C block (2x2 WMMA tiles).
// Double-buffered LDS: next k-slab's global_load_b128 issues before the 16-WMMA burst.
// Batch z decomposes as zo = z / zInner, zi = z % zInner with independent strides (covers (b,h)).
// mode 1: scores epilogue  C = C*scale + causal_mask(row, col).
__global__ __launch_bounds__(256) void gemm_bt_wmma(
    const float* __restrict__ Ag, const float* __restrict__ Bg, float* __restrict__ Cg,
    int M, int N, int K, int lda, int ldb, int ldc,
    long long aOut, long long aIn, long long bOut, long long bIn,
    long long cOut, long long cIn, int zInner,
    int mode, float scale)
{
  __shared__ float As[2][BM * LDSS];
  __shared__ float Bs[2][BN * LDSS];

  const int tid  = threadIdx.x;
  const int lane = tid & 31;
  const int wid  = tid >> 5;
  const int wm   = wid & 3;          // 4 waves along M: 32 rows each (2 tiles)
  const int wn   = wid >> 2;         // 2 waves along N: 32 cols each (2 tiles)
  const int bm   = blockIdx.y * BM;
  const int bn   = blockIdx.x * BN;
  const int z    = blockIdx.z;
  const int zo   = z / zInner, zi = z % zInner;

  const float* A = Ag + zo * aOut + zi * aIn;
  const float* B = Bg + zo * bOut + zi * bIn;
  float*       C = Cg + zo * cOut + zi * cIn;

  // Stagers: float4 i -> row = i>>2, col = (i&3)*4. A: i = tid and tid+256; B: i = tid.
  const int lrow = tid >> 2;         // 0..63
  const int lcol = (tid & 3) << 2;   // 0,4,8,12
  const float* aPtr0 = A + (size_t)(bm + lrow) * lda + lcol;
  const float* aPtr1 = A + (size_t)(bm + 64 + lrow) * lda + lcol;
  const float* bPtr  = B + (size_t)(bn + lrow) * ldb + lcol;

  v8f acc00 = {0,0,0,0,0,0,0,0}, acc01 = {0,0,0,0,0,0,0,0};
  v8f acc10 = {0,0,0,0,0,0,0,0}, acc11 = {0,0,0,0,0,0,0,0};

  // prologue: stage k-slab 0
  float4 a4_0 = *(const float4*)(aPtr0);
  float4 a4_1 = *(const float4*)(aPtr1);
  float4 b4   = *(const float4*)(bPtr);
  *(float4*)&As[0][lrow * LDSS + lcol]        = a4_0;
  *(float4*)&As[0][(64 + lrow) * LDSS + lcol] = a4_1;
  *(float4*)&Bs[0][lrow * LDSS + lcol]        = b4;
  __syncthreads();

  const int nIter = K / BK;
  for (int it = 0; it < nIter; ++it) {
    const int cur = it & 1, nxt = cur ^ 1;
    const int k0  = it * BK;
    const bool more = (it + 1) < nIter;
    if (more) {                       // overlap next slab's loads with WMMA burst
      a4_0 = *(const float4*)(aPtr0 + k0 + BK);
      a4_1 = *(const float4*)(aPtr1 + k0 + BK);
      b4   = *(const float4*)(bPtr  + k0 + BK);
      if (it + 2 < nIter) {
        __builtin_prefetch(aPtr0 + k0 + 2 * BK, 0, 3);
        __builtin_prefetch(aPtr1 + k0 + 2 * BK, 0, 3);
        __builtin_prefetch(bPtr  + k0 + 2 * BK, 0, 3);
      }
    }
#pragma unroll
    for (int kk = 0; kk < BK; kk += 4) {
      const v2f a0 = lds_frag(As[cur], wm * 32,      kk, lane);
      const v2f a1 = lds_frag(As[cur], wm * 32 + 16, kk, lane);
      const v2f b0 = lds_frag(Bs[cur], wn * 32,      kk, lane);
      const v2f b1 = lds_frag(Bs[cur], wn * 32 + 16, kk, lane);
      acc00 = wmma4(a0, b0, acc00);
      acc01 = wmma4(a0, b1, acc01);
      acc10 = wmma4(a1, b0, acc10);
      acc11 = wmma4(a1, b1, acc11);
    }
    if (more) {
      __syncthreads();                // all waves done reading buf[cur] (and buf[nxt] from it-1)
      *(float4*)&As[nxt][lrow * LDSS + lcol]        = a4_0;
      *(float4*)&As[nxt][(64 + lrow) * LDSS + lcol] = a4_1;
      *(float4*)&Bs[nxt][lrow * LDSS + lcol]        = b4;
      __syncthreads();                // stores visible before next iteration's frag reads
    }
  }

  // C/D 16x16 layout: VGPR r -> row (r + 8*(lane>=16)), col = lane&15
  const int cn = lane & 15;
  const int cm = (lane >> 4) << 3;
#pragma unroll
  for (int i = 0; i < 2; ++i) {
    const int gr0 = bm + wm * 32 + i * 16 + cm;
#pragma unroll
    for (int j = 0; j < 2; ++j) {
      const int gc = bn + wn * 32 + j * 16 + cn;
      const v8f acc = (i == 0) ? (j == 0 ? acc00 : acc01) : (j == 0 ? acc10 : acc11);
#pragma unroll
      for (int r = 0; r < 8; ++r) {
        float v = acc[r];
        const int gr = gr0 + r;
        if (mode == 1) v = v * scale + ((gc <= gr) ? 0.0f : -3.0e38f);
        C[(size_t)gr * ldc + gc] = v;
      }
    }
  }
}

// Batched 32x32-tiled transpose: out[z][c][r] = in[z][r][c]
__global__ __launch_bounds__(256) void transpose_batched(
    const float* __restrict__ in, float* __restrict__ out,
    int R, int C, int ldin, long long inBatch, long long outBatch)
{
  __shared__ float tile[32][33];
  const int tx = threadIdx.x & 31;
  const int ty = threadIdx.x >> 5;   // 0..7
  const int r0 = blockIdx.y * 32;
  const int c0 = blockIdx.x * 32;
  const float* ip = in + blockIdx.z * inBatch;
  float*       op = out + blockIdx.z * outBatch;
#pragma unroll
  for (int i = 0; i < 4; ++i)
    tile[ty + 8 * i][tx] = ip[(size_t)(r0 + ty + 8 * i) * ldin + c0 + tx];
  __syncthreads();
#pragma unroll
  for (int i = 0; i < 4; ++i)
    op[(size_t)(c0 + ty + 8 * i) * R + r0 + tx] = tile[tx][ty + 8 * i];
}

// One 256-thread block per row: y = x * rsqrt(mean(x^2)+eps) * w
__global__ __launch_bounds__(256) void rmsnorm_rows(
    const float* __restrict__ in, float* __restrict__ out, const float* __restrict__ w,
    int L, int ldin, int ldout)
{
  __shared__ float red[256];
  const size_t row = blockIdx.x;
  const int tid = threadIdx.x;
  const float* x = in + row * ldin;
  float s = 0.f;
  for (int j = tid; j < L; j += 256) { const float v = x[j]; s += v * v; }
  red[tid] = s; __syncthreads();
  for (int off = 128; off > 0; off >>= 1) {
    if (tid < off) red[tid] += red[tid + off];
    __syncthreads();
  }
  const float inv = rsqrtf(red[0] / (float)L + 1e-6f);
  float* y = out + row * ldout;
  for (int j = tid; j < L; j += 256) y[j] = x[j] * inv * w[j];
}

// RoPE on the 64-dim PE slices; scatters k_pe into kcat[...,512:] and each head's q_pe into qcat[...,512:]
__global__ __launch_bounds__(64) void rope_scatter(
    const float* __restrict__ q,       // (B*S, H*192)
    const float* __restrict__ kvfull,  // (B*S, 576)
    const float* __restrict__ cosp, const float* __restrict__ sinp, // (B*S, 64)
    float* __restrict__ qcat,          // (B, H, S, 576)
    float* __restrict__ kcat)          // (B, S, 576)
{
  using namespace cfg;
  const int row = blockIdx.x;          // b*S + s
  const int j   = threadIdx.x;         // 0..63
  const int b   = row / S, s = row % S;
  const float c  = cosp[(size_t)row * 64 + j];
  const float sn = sinp[(size_t)row * 64 + j];
  {
    const float* x = kvfull + (size_t)row * KC + KVL;
    const float rh = (j < 32) ? -x[j + 32] : x[j - 32];
    kcat[(size_t)row * KC + KVL + j] = x[j] * c + rh * sn;
  }
  for (int h = 0; h < H; ++h) {
    const float* x = q + (size_t)row * (H * QKH) + h * QKH + NOPE;
    const float rh = (j < 32) ? -x[j + 32] : x[j - 32];
    qcat[(((size_t)(b * H + h) * S) + s) * KC + KVL + j] = x[j] * c + rh * sn;
  }
}

// In-place softmax over rows of length 2048 (one block per row, 8 vals/thread held in VGPRs)
__global__ __launch_bounds__(256) void softmax_rows2048(float* __restrict__ p)
{
  __shared__ float red[256];
  float* x = p + (size_t)blockIdx.x * 2048;
  const int tid = threadIdx.x;
  float v[8];
  float m = -3.4e38f;
#pragma unroll
  for (int i = 0; i < 8; ++i) { v[i] = x[tid + 256 * i]; m = fmaxf(m, v[i]); }
  red[tid] = m; __syncthreads();
  for (int off = 128; off > 0; off >>= 1) {
    if (tid < off) red[tid] = fmaxf(red[tid], red[tid + off]);
    __syncthreads();
  }
  m = red[0]; __syncthreads();
  float s = 0.f;
#pragma unroll
  for (int i = 0; i < 8; ++i) { v[i] = expf(v[i] - m); s += v[i]; }
  red[tid] = s; __syncthreads();
  for (int off = 128; off > 0; off >>= 1) {
    if (tid < off) red[tid] += red[tid + off];
    __syncthreads();
  }
  const float inv = 1.0f / red[0];
#pragma unroll
  for (int i = 0; i < 8; ++i) x[tid + 256 * i] = v[i] * inv;
}

extern "C" void kernel_launch(void* const* d_in, const int* in_sizes, int n_in,
                              void* d_out, int out_size, void* d_ws, size_t ws_size,
                              hipStream_t stream)
{
  using namespace cfg;
  (void)in_sizes; (void)n_in; (void)out_size; (void)ws_size;

  const float* hidden  = (const float*)d_in[0];
  const float* cosp    = (const float*)d_in[1];
  const float* sinp    = (const float*)d_in[2];
  /* d_in[3] = attention_mask: causal, recomputed inline in the scores epilogue */
  const float* wq_a    = (const float*)d_in[4];
  const float* q_norm  = (const float*)d_in[5];
  const float* wq_b    = (const float*)d_in[6];
  const float* wkv_a   = (const float*)d_in[7];
  const float* kv_norm = (const float*)d_in[8];
  const float* wkv_b   = (const float*)d_in[9];
  const float* wo      = (const float*)d_in[10];

  float* attn_out = (float*)d_out;                          // (B,S,2048)
  float* attn_w   = attn_out + (size_t)B * S * HIDDEN;      // (B,H,S,S)

  float* ws = (float*)d_ws;
  float* qa     = ws; ws += (size_t)B * S * QL;             // (B*S,768)
  float* qbuf   = ws; ws += (size_t)B * S * H * QKH;        // (B*S,3072)
  float* kvfull = ws; ws += (size_t)B * S * KC;             // (B*S,576)
  float* kcat   = ws; ws += (size_t)B * S * KC;             // (B,S,576)  norm'd kv | rope'd k_pe
  float* kvnT   = ws; ws += (size_t)B * KVL * S;            // (B,512,2048)
  float* wkvbT  = ws; ws += (size_t)H * KVL * NOPE;         // (H,512,128)
  float* qcat   = ws; ws += (size_t)B * H * S * KC;         // (B,H,S,576) q_lat | rope'd q_pe
  float* ctx    = qcat;                                     // reuse: (B,H,S,512) after scores consume qcat
  float* outh   = qbuf;                                     // reuse: (B,S,2048) after q consumed

  const dim3 blk(256);
  const float scaling = 1.0f / sqrtf((float)QKH);

  // T1: W_uk (per-head rows 0..127 of wkv_b) -> (H,512,128) so q_lat GEMM is A*B^T
  transpose_batched<<<dim3(KVL/32, NOPE/32, H), blk, 0, stream>>>(
      wkv_b, wkvbT, NOPE, KVL, KVL, (long long)(NOPE+VD)*KVL, (long long)KVL*NOPE);

  // G1: qa = hidden @ wq_a^T          (4096 x 768 x 2048)
  gemm_bt_wmma<<<dim3(QL/BN, B*S/BM, 1), blk, 0, stream>>>(
      hidden, wq_a, qa, B*S, QL, HIDDEN, HIDDEN, HIDDEN, QL,
      0,0,0,0,0,0, 1, 0, 1.0f);
  rmsnorm_rows<<<dim3(B*S), blk, 0, stream>>>(qa, qa, q_norm, QL, QL, QL);

  // G2: q = qa @ wq_b^T               (4096 x 3072 x 768)
  gemm_bt_wmma<<<dim3(H*QKH/BN, B*S/BM, 1), blk, 0, stream>>>(
      qa, wq_b, qbuf, B*S, H*QKH, QL, QL, QL, H*QKH,
      0,0,0,0,0,0, 1, 0, 1.0f);

  // G3: kvfull = hidden @ wkv_a^T     (4096 x 576 x 2048)
  gemm_bt_wmma<<<dim3(KC/BN, B*S/BM, 1), blk, 0, stream>>>(
      hidden, wkv_a, kvfull, B*S, KC, HIDDEN, HIDDEN, HIDDEN, KC,
      0,0,0,0,0,0, 1, 0, 1.0f);
  rmsnorm_rows<<<dim3(B*S), blk, 0, stream>>>(kvfull, kcat, kv_norm, KVL, KC, KC);

  // RoPE: fills qcat[...,512:576] and kcat[...,512:576]
  rope_scatter<<<dim3(B*S), dim3(64), 0, stream>>>(qbuf, kvfull, cosp, sinp, qcat, kcat);

  // G4: q_lat -> qcat[...,:512]; per (b,h): (2048 x 512 x 128)
  gemm_bt_wmma<<<dim3(KVL/BN, S/BM, B*H), blk, 0, stream>>>(
      qbuf, wkvbT, qcat, S, KVL, NOPE, H*QKH, NOPE, KC,
      (long long)S*H*QKH, QKH, 0, (long long)KVL*NOPE,
      (long long)H*S*KC, (long long)S*KC, H, 0, 1.0f);

  // T2: kv (norm'd) -> (B,512,2048) for the ctx GEMM's B^T form
  transpose_batched<<<dim3(KVL/32, S/32, B), blk, 0, stream>>>(
      kcat, kvnT, S, KVL, KC, (long long)S*KC, (long long)KVL*S);

  // G5: scores = (qcat @ kcat^T)*scale + causal -> attn_w; per (b,h): (2048 x 2048 x 576)
  gemm_bt_wmma<<<dim3(S/BN, S/BM, B*H), blk, 0, stream>>>(
      qcat, kcat, attn_w, S, S, KC, KC, KC, S,
      (long long)H*S*KC, (long long)S*KC,
      (long long)S*KC, 0,
      (long long)H*S*S, (long long)S*S, H, 1, scaling);

  softmax_rows2048<<<dim3(B*H*S), blk, 0, stream>>>(attn_w);

  // G6: ctx = P @ kv; per (b,h): (2048 x 512 x 2048)
  gemm_bt_wmma<<<dim3(KVL/BN, S/BM, B*H), blk, 0, stream>>>(
      attn_w, kvnT, ctx, S, KVL, S, S, S, KVL,
      (long long)H*S*S, (long long)S*S,
      (long long)KVL*S, 0,
      (long long)H*S*KVL, (long long)S*KVL, H, 0, 1.0f);

  // G7: outh[b,s,h*128+d] = ctx @ W_uv^T; per (b,h): (2048 x 128 x 512)
  gemm_bt_wmma<<<dim3(VD/BN, S/BM, B*H), blk, 0, stream>>>(
      ctx, wkv_b + (size_t)NOPE*KVL, outh, S, VD, KVL, KVL, KVL, H*VD,
      (long long)H*S*KVL, (long long)S*KVL,
      0, (long long)(NOPE+VD)*KVL,
      (long long)S*H*VD, VD, H, 0, 1.0f);

  // G8: attn_output = outh @ wo^T     (4096 x 2048 x 2048)
  gemm_bt_wmma<<<dim3(HIDDEN/BN, B*S/BM, 1), blk, 0, stream>>>(
      outh, wo, attn_out, B*S, HIDDEN, H*VD, H*VD, HIDDEN, HIDDEN,
      0,0,0,0,0,0, 1, 0, 1.0f);
}